// CSCOut_61503931678830
// MI455X (gfx1250) — compile-verified
//
#include <hip/hip_runtime.h>
#include <hip/hip_bf16.h>

typedef float v2f __attribute__((ext_vector_type(2)));
typedef float v8f __attribute__((ext_vector_type(8)));

// ======================= constexpr Wigner-3j machinery =======================
struct cpx { double re, im; };
constexpr cpx cmul(cpx a, cpx b){ return cpx{a.re*b.re - a.im*b.im, a.re*b.im + a.im*b.re}; }

constexpr double csqrt(double x){
  if (x <= 0.0) return 0.0;
  double g = x > 1.0 ? x : 1.0;
  for (int i = 0; i < 100; ++i) g = 0.5*(g + x/g);
  return g;
}
constexpr double cfact(int n){ double r = 1.0; for (int i = 2; i <= n; ++i) r *= (double)i; return r; }
constexpr int imax3(int a,int b,int c){ int m = a > b ? a : b; return m > c ? m : c; }
constexpr int imin3(int a,int b,int c){ int m = a < b ? a : b; return m < c ? m : c; }

constexpr double cgc(int j1,int m1,int j2,int m2,int j3,int m3){
  if (m3 != m1 + m2) return 0.0;
  const int vmin = imax3(-j1 + j2 + m3, -j1 + m1, 0);
  const int vmax = imin3(j2 + j3 + m1, j3 - j1 + j2, j3 + m3);
  double pref = csqrt((2.0*j3 + 1.0)*cfact(j3 + j1 - j2)*cfact(j3 - j1 + j2)*cfact(j1 + j2 - j3)/cfact(j1 + j2 + j3 + 1));
  pref *= csqrt(cfact(j3 + m3)*cfact(j3 - m3)/(cfact(j1 + m1)*cfact(j1 - m1)*cfact(j2 + m2)*cfact(j2 - m2)));
  double s = 0.0;
  for (int v = vmin; v <= vmax; ++v){
    const double sgn = ((v + j2 + m2) & 1) ? -1.0 : 1.0;
    s += sgn * cfact(j2 + j3 + m1 - v) * cfact(j1 - m1 + v)
         / (cfact(v) * cfact(j3 - j1 + j2 - v) * cfact(j3 + m3 - v) * cfact(v + j1 - j2 - m3));
  }
  return pref * s;
}

template<int L> struct R2C { cpx q[2*L+1][2*L+1]; };

template<int L>
constexpr R2C<L> make_r2c(){
  R2C<L> r{};
  const double s2 = 0.70710678118654752440;
  for (int m = -L; m < 0; ++m){
    r.q[L+m][L-m] = cpx{s2, 0.0};      // q[l+m, l+|m|]
    r.q[L+m][L+m] = cpx{0.0, -s2};     // q[l+m, l-|m|]
  }
  r.q[L][L] = cpx{1.0, 0.0};
  for (int m = 1; m <= L; ++m){
    const double sg = (m & 1) ? -1.0 : 1.0;
    r.q[L+m][L+m] = cpx{sg*s2, 0.0};
    r.q[L+m][L-m] = cpx{0.0, sg*s2};
  }
  const cpx f = (L == 0) ? cpx{1.0, 0.0} : ((L == 1) ? cpx{0.0, -1.0} : cpx{-1.0, 0.0}); // (-i)^L
  for (int i = 0; i < 2*L+1; ++i)
    for (int j = 0; j < 2*L+1; ++j)
      r.q[i][j] = cmul(f, r.q[i][j]);
  return r;
}

template<int L1,int L2,int L3> struct W3J { float v[2*L1+1][2*L2+1][2*L3+1]; };

template<int L1,int L2,int L3>
constexpr W3J<L1,L2,L3> make_w3j(){
  constexpr int D1 = 2*L1+1, D2 = 2*L2+1, D3 = 2*L3+1;
  double C[D1][D2][D3] = {};
  const double inv = 1.0/csqrt(2.0*L3 + 1.0);
  for (int m1 = -L1; m1 <= L1; ++m1)
    for (int m2 = -L2; m2 <= L2; ++m2){
      const int m3 = m1 + m2;
      if (m3 >= -L3 && m3 <= L3)
        C[L1+m1][L2+m2][L3+m3] = cgc(L1,m1,L2,m2,L3,m3)*inv;
    }
  const R2C<L1> q1 = make_r2c<L1>();
  const R2C<L2> q2 = make_r2c<L2>();
  const R2C<L3> q3 = make_r2c<L3>();
  double re[D1][D2][D3] = {};
  double im[D1][D2][D3] = {};
  for (int j = 0; j < D1; ++j)
    for (int l = 0; l < D2; ++l)
      for (int n = 0; n < D3; ++n){
        double ar = 0.0, ai = 0.0;
        for (int i = 0; i < D1; ++i)
          for (int k = 0; k < D2; ++k)
            for (int m = 0; m < D3; ++m){
              const double c = C[i][k][m];
              if (c != 0.0){
                cpx t = cmul(q1.q[i][j], q2.q[k][l]);
                t = cmul(t, q3.q[m][n]);
                ar += t.re*c; ai += t.im*c;
              }
            }
        re[j][l][n] = ar; im[j][l][n] = ai;
      }
  double nr = 0.0, ni = 0.0;
  for (int j = 0; j < D1; ++j) for (int l = 0; l < D2; ++l) for (int n = 0; n < D3; ++n){
    nr += re[j][l][n]*re[j][l][n];
    ni += im[j][l][n]*im[j][l][n];
  }
  const bool useIm = ni > nr;
  const double nrm = csqrt(useIm ? ni : nr);
  W3J<L1,L2,L3> out{};
  for (int j = 0; j < D1; ++j) for (int l = 0; l < D2; ++l) for (int n = 0; n < D3; ++n)
    out.v[j][l][n] = (float)((useIm ? im[j][l][n] : re[j][l][n]) / nrm);
  return out;
}

constexpr W3J<0,0,0> C000 = make_w3j<0,0,0>();
constexpr W3J<1,1,0> C110 = make_w3j<1,1,0>();
constexpr W3J<2,2,0> C220 = make_w3j<2,2,0>();
constexpr W3J<1,1,1> C111 = make_w3j<1,1,1>();
constexpr W3J<2,2,1> C221 = make_w3j<2,2,1>();
constexpr W3J<0,2,2> C022 = make_w3j<0,2,2>();
constexpr W3J<1,1,2> C112 = make_w3j<1,1,2>();
constexpr W3J<2,0,2> C202 = make_w3j<2,0,2>();
constexpr W3J<2,2,2> C222 = make_w3j<2,2,2>();

struct QP9 { float v[9][3][3]; };
constexpr QP9 make_QP(){
  QP9 q{};
  const W3J<1,1,0> w0 = make_w3j<1,1,0>();
  const W3J<1,1,1> w1 = make_w3j<1,1,1>();
  const W3J<1,1,2> w2 = make_w3j<1,1,2>();
  const double r3 = csqrt(3.0), r5 = csqrt(5.0);
  const int P[3] = {2,0,1};                       // res[:, idx, :][:, :, idx] fold
  for (int i = 0; i < 3; ++i)
    for (int j = 0; j < 3; ++j){
      const int pi = P[i], pj = P[j];
      q.v[0][i][j] = w0.v[pi][pj][0];
      for (int k = 0; k < 3; ++k) q.v[1+k][i][j] = (float)(r3 * (double)w1.v[pi][pj][k]);
      for (int k = 0; k < 5; ++k) q.v[4+k][i][j] = (float)(r5 * (double)w2.v[pi][pj][k]);
    }
  return q;
}
constexpr QP9 QP = make_QP();

constexpr float N0c = 0.14433756729740643f;  // sqrt(1/48)
constexpr float N1c = 0.30618621784789724f;  // sqrt(3/32)
constexpr float N2c = 0.27950849718747373f;  // sqrt(5/64)

// ======================= device helpers =======================
__device__ __forceinline__ v8f wmma4(v2f a, v2f b, v8f c){
  // V_WMMA_F32_16X16X4_F32 : D = A(16x4,f32) x B(4x16,f32) + C(16x16,f32)
  return __builtin_amdgcn_wmma_f32_16x16x4_f32(false, a, false, b, (short)0, c, false, false);
}

template<int D1,int D2,int D3>
__device__ __forceinline__ void tp_acc(const float (&C)[D1][D2][D3],
                                       float wgt, const float* x1, const float* x2,
                                       float nrm, float* o)
{
  #pragma unroll
  for (int k = 0; k < D3; ++k){
    float acc = 0.f;
    #pragma unroll
    for (int i = 0; i < D1; ++i){
      #pragma unroll
      for (int j = 0; j < D2; ++j){
        const float c = C[i][j][k];
        if (c != 0.f) acc += c * x1[i] * x2[j];   // constant-folded; zero terms vanish
      }
    }
    o[k] += nrm * wgt * acc;
  }
}

#define WAVES_PER_BLOCK 2
#define ATOM_STRIDE 289   // 289 % 64 == 33, coprime with banks -> conflict-free
#define OFF_H 0           // h[64]   (overwritten by s/v after w-GEMM)
#define OFF_S 0           // s[16]
#define OFF_V 16          // v[16][3]
#define OFF_T 64          // t[16][5]
#define OFF_W 144         // w[9][16]

// ======================= kernel =======================
extern "C" __global__ void __launch_bounds__(64)
csc_equiv_kernel(const float* __restrict__ xs,   // (N,128)
                 const float* __restrict__ xsph, // (N,480)
                 const float* __restrict__ Wl0,  // (128,16)
                 const float* __restrict__ Wl1,  // (64,16)
                 const float* __restrict__ Wl2,  // (32,16)
                 const float* __restrict__ Wm1,  // (128,64)
                 const float* __restrict__ bm1,  // (64)
                 const float* __restrict__ Wm2,  // (64,144)
                 const float* __restrict__ bm2,  // (144)
                 float* __restrict__ out,        // (N,3,3)
                 int N, int nTiles)
{
  __shared__ float smem[WAVES_PER_BLOCK * 16 * ATOM_STRIDE];
  const int wave = threadIdx.x >> 5;
  const int lane = threadIdx.x & 31;
  const int half = lane >> 4;     // K-half for A/B fragments
  const int lrow = lane & 15;     // A row / D column
  const int tile = blockIdx.x * WAVES_PER_BLOCK + wave;
  if (tile >= nTiles) return;     // wave-uniform: EXEC stays all-ones for WMMA

  const int atom0 = tile * 16;
  int arow = atom0 + lrow; if (arow >= N) arow = N - 1;   // clamp tail reads
  float* wb = smem + wave * 16 * ATOM_STRIDE;

  // ---------------- h = silu(xs @ Wm1 + b1)  [M16,N64,K128] ----------------
  {
    v8f acc[4];
    #pragma unroll
    for (int nt = 0; nt < 4; ++nt) acc[nt] = v8f{0.f,0.f,0.f,0.f,0.f,0.f,0.f,0.f};
    const float* ap = xs + (size_t)arow * 128;
    #pragma unroll 4
    for (int st = 0; st < 32; ++st){
      const int k = 4*st + 2*half;
      const v2f a = *(const v2f*)(ap + k);
      #pragma unroll
      for (int nt = 0; nt < 4; ++nt){
        v2f b; b.x = Wm1[k*64 + nt*16 + lrow]; b.y = Wm1[(k+1)*64 + nt*16 + lrow];
        acc[nt] = wmma4(a, b, acc[nt]);
      }
    }
    #pragma unroll
    for (int nt = 0; nt < 4; ++nt){
      const float bias = bm1[nt*16 + lrow];
      #pragma unroll
      for (int r = 0; r < 8; ++r){
        const float x = acc[nt][r] + bias;
        const float h = x / (1.f + __expf(-x));          // silu
        wb[(r + 8*half)*ATOM_STRIDE + OFF_H + nt*16 + lrow] = h;
      }
    }
  }

  // ---------------- w = h @ Wm2 + b2  [M16,N144,K64], A from LDS -----------
  #pragma unroll 1
  for (int p = 0; p < 9; ++p){
    v8f acc = v8f{0.f,0.f,0.f,0.f,0.f,0.f,0.f,0.f};
    #pragma unroll 4
    for (int st = 0; st < 16; ++st){
      const int k = 4*st + 2*half;
      v2f a; a.x = wb[lrow*ATOM_STRIDE + OFF_H + k];
             a.y = wb[lrow*ATOM_STRIDE + OFF_H + k + 1];
      v2f b; b.x = Wm2[k*144 + p*16 + lrow]; b.y = Wm2[(k+1)*144 + p*16 + lrow];
      acc = wmma4(a, b, acc);
    }
    const float bias = bm2[p*16 + lrow];
    #pragma unroll
    for (int r = 0; r < 8; ++r)
      wb[(r + 8*half)*ATOM_STRIDE + OFF_W + p*16 + lrow] = acc[r] + bias;
  }

  // ---------------- s = s_in @ Wl0 / sqrt(128)  [M16,N16,K128] -------------
  {
    v8f acc = v8f{0.f,0.f,0.f,0.f,0.f,0.f,0.f,0.f};
    const float* ap = xsph + (size_t)arow * 480;
    #pragma unroll 4
    for (int st = 0; st < 32; ++st){
      const int k = 4*st + 2*half;
      const v2f a = *(const v2f*)(ap + k);
      v2f b; b.x = Wl0[k*16 + lrow]; b.y = Wl0[(k+1)*16 + lrow];
      acc = wmma4(a, b, acc);
    }
    const float sc = 0.088388347648318447f;  // 1/sqrt(128)
    #pragma unroll
    for (int r = 0; r < 8; ++r)
      wb[(r + 8*half)*ATOM_STRIDE + OFF_S + lrow] = acc[r]*sc;
  }

  // ---------------- v[c] = v_in[:,:,c] @ Wl1 / sqrt(64)  ------------------
  #pragma unroll 1
  for (int c = 0; c < 3; ++c){
    v8f acc = v8f{0.f,0.f,0.f,0.f,0.f,0.f,0.f,0.f};
    const float* ap = xsph + (size_t)arow * 480 + 128 + c;  // stride 3 over K
    #pragma unroll 4
    for (int st = 0; st < 16; ++st){
      const int k = 4*st + 2*half;
      v2f a; a.x = ap[k*3]; a.y = ap[(k+1)*3];
      v2f b; b.x = Wl1[k*16 + lrow]; b.y = Wl1[(k+1)*16 + lrow];
      acc = wmma4(a, b, acc);
    }
    const float sc = 0.125f;                 // 1/sqrt(64)
    #pragma unroll
    for (int r = 0; r < 8; ++r)
      wb[(r + 8*half)*ATOM_STRIDE + OFF_V + lrow*3 + c] = acc[r]*sc;
  }

  // ---------------- t[c] = t_in[:,:,c] @ Wl2 / sqrt(32) -------------------
  #pragma unroll 1
  for (int c = 0; c < 5; ++c){
    v8f acc = v8f{0.f,0.f,0.f,0.f,0.f,0.f,0.f,0.f};
    const float* ap = xsph + (size_t)arow * 480 + 320 + c;  // stride 5 over K
    #pragma unroll 4
    for (int st = 0; st < 8; ++st){
      const int k = 4*st + 2*half;
      v2f a; a.x = ap[k*5]; a.y = ap[(k+1)*5];
      v2f b; b.x = Wl2[k*16 + lrow]; b.y = Wl2[(k+1)*16 + lrow];
      acc = wmma4(a, b, acc);
    }
    const float sc = 0.17677669529663687f;   // 1/sqrt(32)
    #pragma unroll
    for (int r = 0; r < 8; ++r)
      wb[(r + 8*half)*ATOM_STRIDE + OFF_T + lrow*5 + c] = acc[r]*sc;
  }

  // ---------------- phase 2: tensor products + Q projection ----------------
  if (lane < 16){
    const int atom = atom0 + lrow;
    const float* ap = wb + lrow*ATOM_STRIDE;
    float o[9] = {0.f,0.f,0.f,0.f,0.f,0.f,0.f,0.f,0.f};
    for (int u = 0; u < 16; ++u){
      float su[1]; su[0] = ap[OFF_S + u];
      float vu[3];
      #pragma unroll
      for (int i = 0; i < 3; ++i) vu[i] = ap[OFF_V + u*3 + i];
      float tu[5];
      #pragma unroll
      for (int i = 0; i < 5; ++i) tu[i] = ap[OFF_T + u*5 + i];
      float wv[9];
      #pragma unroll
      for (int p = 0; p < 9; ++p) wv[p] = ap[OFF_W + p*16 + u];

      tp_acc(C000.v, wv[0], su, su, N0c, &o[0]);
      tp_acc(C110.v, wv[1], vu, vu, N0c, &o[0]);
      tp_acc(C220.v, wv[2], tu, tu, N0c, &o[0]);
      tp_acc(C111.v, wv[3], vu, vu, N1c, &o[1]);
      tp_acc(C221.v, wv[4], tu, tu, N1c, &o[1]);
      tp_acc(C022.v, wv[5], su, tu, N2c, &o[4]);
      tp_acc(C112.v, wv[6], vu, vu, N2c, &o[4]);
      tp_acc(C202.v, wv[7], tu, su, N2c, &o[4]);
      tp_acc(C222.v, wv[8], tu, tu, N2c, &o[4]);
    }
    if (atom < N){
      float* op = out + (size_t)atom * 9;
      #pragma unroll
      for (int i = 0; i < 3; ++i){
        #pragma unroll
        for (int j = 0; j < 3; ++j){
          float r = 0.f;
          #pragma unroll
          for (int a = 0; a < 9; ++a){
            const float q = QP.v[a][i][j];
            if (q != 0.f) r += o[a]*q;
          }
          op[i*3 + j] = r;
        }
      }
    }
  }
}

// ======================= launch =======================
extern "C" void kernel_launch(void* const* d_in, const int* in_sizes, int n_in,
                              void* d_out, int out_size, void* d_ws, size_t ws_size,
                              hipStream_t stream) {
  (void)n_in; (void)out_size; (void)d_ws; (void)ws_size;
  const float* xs   = (const float*)d_in[0];
  const float* xsph = (const float*)d_in[1];
  // d_in[2] = coord (unused), d_in[3] = batch_index (unused)
  const float* Wl0  = (const float*)d_in[4];
  const float* Wl1  = (const float*)d_in[5];
  const float* Wl2  = (const float*)d_in[6];
  const float* Wm1  = (const float*)d_in[7];
  const float* bm1  = (const float*)d_in[8];
  const float* Wm2  = (const float*)d_in[9];
  const float* bm2  = (const float*)d_in[10];
  float* out = (float*)d_out;

  const int N = in_sizes[0] / 128;
  const int nTiles = (N + 15) / 16;
  const int blocks = (nTiles + WAVES_PER_BLOCK - 1) / WAVES_PER_BLOCK;
  if (blocks > 0) {
    hipLaunchKernelGGL(csc_equiv_kernel, dim3(blocks), dim3(64), 0, stream,
                       xs, xsph, Wl0, Wl1, Wl2, Wm1, bm1, Wm2, bm2, out, N, nTiles);
  }
}